// GNNClassifier_28209345200794
// MI455X (gfx1250) — compile-verified
//
#include <hip/hip_runtime.h>

typedef float v2f __attribute__((ext_vector_type(2)));
typedef float v8f __attribute__((ext_vector_type(8)));

#define BN_EPS 1e-5f

// ---------------------------------------------------------------- utilities
__global__ void fill_kernel(float* __restrict__ p, long n, float val) {
    long i = (long)blockIdx.x * blockDim.x + threadIdx.x;
    if (i < n) p[i] = val;
}

__device__ __forceinline__ void atomicAddF(float* p, float v) {
    // lowers to global_atomic_add_f32 (no CAS loop)
    unsafeAtomicAdd(p, v);
}

// degree: deg[] pre-filled with 1.0 (self loop), count incoming edges
__global__ void deg_count_kernel(const int* __restrict__ dst, float* __restrict__ deg, int e) {
    int i = blockIdx.x * blockDim.x + threadIdx.x;
    if (i < e) atomicAddF(&deg[dst[i]], 1.0f);
}

__global__ void dinv_kernel(float* __restrict__ deg, int n) {
    int i = blockIdx.x * blockDim.x + threadIdx.x;
    if (i < n) {
        float d = deg[i];
        deg[i] = d > 0.0f ? rsqrtf(d) : 0.0f;
    }
}

// ------------------------------------------------------------- WMMA f32 GEMM
// C[M,H] = A[M,K] @ W[K,H], row-major f32.  8 waves / block; each wave owns a
// 16x64 output strip (4 accumulators); all waves share a 64-column slab of W
// staged in LDS in *fragment* layout:
//   ldsW[p*STRIDE + c] = float2( W[2p][cbase+c], W[2p+1][cbase+c] )
// so a lane's B fragment for V_WMMA_F32_16X16X4_F32 is one ds_load_b64:
//   lane l, k-step k:  p = k/2 + (l>>4), c = tile*16 + (l&15)
// STRIDE = 64+16 pads the two half-waves onto disjoint 32-bank halves.
//
// V_WMMA_F32_16X16X4_F32 operand layouts (ISA 7.12.2):
//   A (16x4):  lane l, vgpr v ->  M = l&15, K = 2*(l>>4) + v   (float2 load)
//   B (4x16):  lane l, vgpr v ->  N = l&15, K = v + 2*(l>>4)
//   C (16x16): lane l, vgpr i ->  N = l&15, M = i + 8*(l>>4)
template <int NT>   // NT = tiles per wave in H; slab width = 16*NT
__global__ void wmma_gemm_f32(const float* __restrict__ A, const float* __restrict__ W,
                              float* __restrict__ C, int Mtiles, int K, int H) {
    constexpr int COLS   = 16 * NT;
    constexpr int STRIDE = COLS + 16;     // bank-conflict padding
    extern __shared__ v2f ldsW[];         // (K/2) * STRIDE float2

    const int tid   = threadIdx.x;
    const int lane  = tid & 31;
    const int wave  = tid >> 5;
    const int l16   = lane & 15;
    const int khalf = lane >> 4;
    const int cbase = blockIdx.x * COLS;

    // ---- cooperative stage of the W slab
    const int npairs = K >> 1;
    for (int idx = tid; idx < npairs * COLS; idx += (int)blockDim.x) {
        int p = idx / COLS;
        int c = idx - p * COLS;
        v2f w2;
        w2.x = W[(long)(2 * p) * H + cbase + c];
        w2.y = W[(long)(2 * p + 1) * H + cbase + c];
        ldsW[p * STRIDE + c] = w2;
    }
    __syncthreads();

    const int mtile = blockIdx.y * ((int)blockDim.x >> 5) + wave;
    if (mtile >= Mtiles) return;          // whole wave exits together

    v8f acc[NT];
#pragma unroll
    for (int t = 0; t < NT; ++t) acc[t] = (v8f){0.f, 0.f, 0.f, 0.f, 0.f, 0.f, 0.f, 0.f};

    const float* arow = A + ((long)mtile * 16 + l16) * K;
#pragma unroll 2
    for (int k = 0; k < K; k += 4) {
        v2f a = *(const v2f*)(arow + k + 2 * khalf);          // streaming, coalesced
        const v2f* bp = ldsW + ((k >> 1) + khalf) * STRIDE + l16;
#pragma unroll
        for (int t = 0; t < NT; ++t) {
            v2f b = bp[t * 16];                                // ds_load_b64
            acc[t] = __builtin_amdgcn_wmma_f32_16x16x4_f32(
                false, a, false, b, (short)0, acc[t], false, false);
        }
    }
#pragma unroll
    for (int t = 0; t < NT; ++t) {
        float* crow = C + ((long)mtile * 16 + 8 * khalf) * H + cbase + t * 16 + l16;
#pragma unroll
        for (int i = 0; i < 8; ++i) crow[(long)i * H] = acc[t][i];
    }
}

// -------------------------------------------------- edge gather + scatter-add
// out[dst] += h[src] * dinv[src]*dinv[dst].  One wave per edge; lane gathers a
// float4/float2 chunk (b128/b64 coalesced) and scatters with f32 atomics.
// Virtual edges [n_edges, n_edges+n_nodes) are the self loops.
template <int VEC>  // C = 32*VEC channels
__global__ void aggregate_kernel(const float* __restrict__ h, const int* __restrict__ src,
                                 const int* __restrict__ dst, const float* __restrict__ dinv,
                                 float* __restrict__ out, int n_edges, int n_nodes) {
    constexpr int C = 32 * VEC;
    long wid = ((long)blockIdx.x * blockDim.x + threadIdx.x) >> 5;
    int lane = threadIdx.x & 31;
    long total = (long)n_edges + n_nodes;
    if (wid >= total) return;             // wave-uniform exit
    int s, d;
    if (wid < n_edges) {
        s = src[wid];
        d = dst[wid];
    } else {
        s = (int)(wid - n_edges);
        d = s;
    }
    s = __builtin_amdgcn_readfirstlane(s);     // force scalar addressing
    d = __builtin_amdgcn_readfirstlane(d);
    float nrm = dinv[s] * dinv[d];
    const float* hs = h + (long)s * C + lane * VEC;
    float* od = out + (long)d * C + lane * VEC;
    if constexpr (VEC == 4) {
        float4 v = *(const float4*)hs;
        atomicAddF(od + 0, v.x * nrm);
        atomicAddF(od + 1, v.y * nrm);
        atomicAddF(od + 2, v.z * nrm);
        atomicAddF(od + 3, v.w * nrm);
    } else {
        float2 v = *(const float2*)hs;
        atomicAddF(od + 0, v.x * nrm);
        atomicAddF(od + 1, v.y * nrm);
    }
}

// --------------------------------------------- fused bias + batchnorm + relu
__global__ void bias_bn_relu_kernel(float* __restrict__ h, const float* __restrict__ b,
                                    const float* __restrict__ g, const float* __restrict__ be,
                                    const float* __restrict__ m, const float* __restrict__ v,
                                    long n, int C) {
    long i = (long)blockIdx.x * blockDim.x + threadIdx.x;
    if (i >= n) return;
    int c = (int)(i % C);
    float val = h[i] + b[c];
    val = (val - m[c]) * (g[c] * rsqrtf(v[c] + BN_EPS)) + be[c];
    h[i] = fmaxf(val, 0.0f);
}

// ------------------------------------------------------------------- pooling
// wave per node, C = 64: lane handles 2 channels (b64 gather + 2 atomics)
__global__ void pool_sum_kernel(const float* __restrict__ h, const int* __restrict__ batch,
                                float* __restrict__ sums, int n) {
    long wid = ((long)blockIdx.x * blockDim.x + threadIdx.x) >> 5;
    int lane = threadIdx.x & 31;
    if (wid >= n) return;
    int g = __builtin_amdgcn_readfirstlane(batch[wid]);
    float2 v = *(const float2*)(h + wid * 64 + lane * 2);
    atomicAddF(&sums[(long)g * 64 + lane * 2 + 0], v.x);
    atomicAddF(&sums[(long)g * 64 + lane * 2 + 1], v.y);
}

__global__ void pool_cnt_kernel(const int* __restrict__ batch, float* __restrict__ cnt, int n) {
    int i = blockIdx.x * blockDim.x + threadIdx.x;
    if (i < n) atomicAddF(&cnt[batch[i]], 1.0f);
}

__global__ void pool_div_kernel(float* __restrict__ sums, const float* __restrict__ cnt,
                                int g, int C) {
    int i = blockIdx.x * blockDim.x + threadIdx.x;
    if (i < g * C) sums[i] /= fmaxf(cnt[i / C], 1.0f);
}

// ------------------------------------------------------------------ MLP tail
__global__ void bias_relu_kernel(float* __restrict__ z, const float* __restrict__ b,
                                 long n, int C) {
    long i = (long)blockIdx.x * blockDim.x + threadIdx.x;
    if (i < n) z[i] = fmaxf(z[i] + b[i % C], 0.0f);
}

__global__ void head_kernel(const float* __restrict__ z, const float* __restrict__ W,
                            const float* __restrict__ b, float* __restrict__ out,
                            int G, int K, int O) {
    int i = blockIdx.x * blockDim.x + threadIdx.x;
    if (i >= G * O) return;
    int g = i / O, j = i % O;
    float s = b[j];
    for (int k = 0; k < K; ++k) s += z[(long)g * K + k] * W[(long)k * O + j];
    out[i] = s;
}

// =========================================================================
extern "C" void kernel_launch(void* const* d_in, const int* in_sizes, int n_in,
                              void* d_out, int out_size, void* d_ws, size_t ws_size,
                              hipStream_t stream) {
    const float* x    = (const float*)d_in[0];
    const int*   edge = (const int*)d_in[1];
    const int*   batch= (const int*)d_in[2];
    const float* W1 = (const float*)d_in[3];  const float* b1 = (const float*)d_in[4];
    const float* W2 = (const float*)d_in[5];  const float* b2 = (const float*)d_in[6];
    const float* W3 = (const float*)d_in[7];  const float* b3 = (const float*)d_in[8];
    const float* g1 = (const float*)d_in[9];  const float* be1= (const float*)d_in[10];
    const float* m1 = (const float*)d_in[11]; const float* v1 = (const float*)d_in[12];
    const float* g2 = (const float*)d_in[13]; const float* be2= (const float*)d_in[14];
    const float* m2 = (const float*)d_in[15]; const float* v2 = (const float*)d_in[16];
    const float* g3 = (const float*)d_in[17]; const float* be3= (const float*)d_in[18];
    const float* m3 = (const float*)d_in[19]; const float* v3 = (const float*)d_in[20];
    const float* Wm1= (const float*)d_in[21]; const float* bm1= (const float*)d_in[22];
    const float* Wm2= (const float*)d_in[23]; const float* bm2= (const float*)d_in[24];

    const int N  = in_sizes[2];       // 100000 nodes (multiple of 16)
    const int E  = in_sizes[1] / 2;   // 1600000 edges
    const int INC = 128, H1c = 128, H2c = 64, G = 128, NC = 10;
    const int* src  = edge;           // edge_index[0]
    const int* dstv = edge + E;       // edge_index[1]

    // ---- workspace carve-out (256B aligned)
    char* ws = (char*)d_ws;
    size_t off = 0;
    auto carve = [&](size_t bytes) -> float* {
        float* p = (float*)(ws + off);
        off += (bytes + 255) & ~(size_t)255;
        return p;
    };
    float* dinv = carve((size_t)N * 4);              // degree -> 1/sqrt(deg)
    float* bufA = carve((size_t)N * H1c * 4);        // transformed features
    float* bufB = carve((size_t)N * H1c * 4);        // aggregated features
    float* psum = carve((size_t)G * H2c * 4);
    float* pcnt = carve((size_t)G * 4);
    float* z1   = carve((size_t)G * H2c * 4);
    (void)n_in; (void)ws_size; (void)out_size;

    const long NE1 = (long)N * H1c;
    const long NE2 = (long)N * H2c;
    const long aggThreads = ((long)E + N) * 32;
    const int  aggBlocks  = (int)((aggThreads + 255) / 256);

    auto gemm = [&](const float* A, const float* Wm, float* Cc, int M, int K, int H) {
        int mtiles = M / 16;
        dim3 grid(H / 64, (mtiles + 7) / 8);
        size_t shmem = (size_t)(K / 2) * 80 * sizeof(v2f);   // STRIDE = 80
        wmma_gemm_f32<4><<<grid, 256, shmem, stream>>>(A, Wm, Cc, mtiles, K, H);
    };

    // ---- degree / normalization coefficients
    fill_kernel<<<(N + 255) / 256, 256, 0, stream>>>(dinv, N, 1.0f);   // self-loop
    deg_count_kernel<<<(E + 255) / 256, 256, 0, stream>>>(dstv, dinv, E);
    dinv_kernel<<<(N + 255) / 256, 256, 0, stream>>>(dinv, N);

    // ---- layer 1: x[N,128] @ W1[128,128] -> aggregate -> BN/ReLU
    gemm(x, W1, bufA, N, INC, H1c);
    fill_kernel<<<(int)((NE1 + 255) / 256), 256, 0, stream>>>(bufB, NE1, 0.0f);
    aggregate_kernel<4><<<aggBlocks, 256, 0, stream>>>(bufA, src, dstv, dinv, bufB, E, N);
    bias_bn_relu_kernel<<<(int)((NE1 + 255) / 256), 256, 0, stream>>>(
        bufB, b1, g1, be1, m1, v1, NE1, H1c);

    // ---- layer 2: h1[N,128] @ W2[128,64]
    gemm(bufB, W2, bufA, N, H1c, H2c);
    fill_kernel<<<(int)((NE2 + 255) / 256), 256, 0, stream>>>(bufB, NE2, 0.0f);
    aggregate_kernel<2><<<aggBlocks, 256, 0, stream>>>(bufA, src, dstv, dinv, bufB, E, N);
    bias_bn_relu_kernel<<<(int)((NE2 + 255) / 256), 256, 0, stream>>>(
        bufB, b2, g2, be2, m2, v2, NE2, H2c);

    // ---- layer 3: h2[N,64] @ W3[64,64]
    gemm(bufB, W3, bufA, N, H2c, H2c);
    fill_kernel<<<(int)((NE2 + 255) / 256), 256, 0, stream>>>(bufB, NE2, 0.0f);
    aggregate_kernel<2><<<aggBlocks, 256, 0, stream>>>(bufA, src, dstv, dinv, bufB, E, N);
    bias_bn_relu_kernel<<<(int)((NE2 + 255) / 256), 256, 0, stream>>>(
        bufB, b3, g3, be3, m3, v3, NE2, H2c);

    // ---- global mean pool
    fill_kernel<<<(G * H2c + 255) / 256, 256, 0, stream>>>(psum, G * H2c, 0.0f);
    fill_kernel<<<1, 256, 0, stream>>>(pcnt, G, 0.0f);
    pool_sum_kernel<<<(int)(((long)N * 32 + 255) / 256), 256, 0, stream>>>(bufB, batch, psum, N);
    pool_cnt_kernel<<<(N + 255) / 256, 256, 0, stream>>>(batch, pcnt, N);
    pool_div_kernel<<<(G * H2c + 255) / 256, 256, 0, stream>>>(psum, pcnt, G, H2c);

    // ---- MLP head: pooled[128,64] @ Wm1[64,64] (+bias, relu), then 64->10
    gemm(psum, Wm1, z1, G, H2c, H2c);
    bias_relu_kernel<<<(G * H2c + 255) / 256, 256, 0, stream>>>(z1, bm1, (long)G * H2c, H2c);
    head_kernel<<<(G * NC + 255) / 256, 256, 0, stream>>>(z1, Wm2, bm2, (float*)d_out, G, H2c, NC);
}